// LocalSelfAttention_68676527063660
// MI455X (gfx1250) — compile-verified
//
#include <hip/hip_runtime.h>
#include <math.h>

typedef float v2f __attribute__((ext_vector_type(2)));
typedef float v8f __attribute__((ext_vector_type(8)));

#define BQ      2
#define SEQ     4096
#define DMODEL  512
#define HEADS   8
#define DH      64
#define WIN     31
#define HALFW   15

// ---------------------------------------------------------------------------
// Projection GEMM: P = X @ W   (X: [8192,512] row-major, W: [512,512])
// fp32 WMMA 16x16x4 (native CDNA5 fp32 matrix op — exact vs reference).
// Block = 256 threads (8 waves). Block tile: 128 rows x 64 cols.
// Wave w computes rows [w*16, w*16+16) x all 64 cols -> 4 v8f accumulators.
// K staged in LDS in blocks of 32; W tile stored TRANSPOSED so each B
// fragment is one aligned ds_load_b64 (no VGPR repacking).
// ---------------------------------------------------------------------------
__global__ __launch_bounds__(256)
void proj_gemm_kernel(const float* __restrict__ X,
                      const float* __restrict__ Wq,
                      const float* __restrict__ Wk,
                      const float* __restrict__ Wv,
                      float* __restrict__ Q,
                      float* __restrict__ Kp,
                      float* __restrict__ Vp)
{
    const float* Wsel;
    float*       Osel;
    if (blockIdx.z == 0)      { Wsel = Wq; Osel = Q;  }
    else if (blockIdx.z == 1) { Wsel = Wk; Osel = Kp; }
    else                      { Wsel = Wv; Osel = Vp; }

    // xs: row stride 36 floats (144B, 8B-aligned; 36*l mod 64 conflict-free)
    // wst: TRANSPOSED W tile [n][k]; row stride 38 floats (152B, 8B-aligned;
    //      38*c mod 64 = 2*(19c mod 32), 19 coprime 32 -> conflict-free writes)
    __shared__ float xs[128][36];
    __shared__ float wst[64][38];

    const int rowBase = blockIdx.x * 128;
    const int colBase = blockIdx.y * 64;
    const int tid     = threadIdx.x;
    const int wave    = tid >> 5;
    const int lane    = tid & 31;
    const int l15     = lane & 15;
    const int koff    = (lane >> 4) << 1;   // lanes 16-31 hold K+2,K+3
    const int waveRow = wave * 16;

    v8f acc0 = {}; v8f acc1 = {}; v8f acc2 = {}; v8f acc3 = {};

    for (int kb = 0; kb < DMODEL; kb += 32) {
        __syncthreads();
        // stage X tile: 128x32 floats (coalesced global, conflict-free LDS)
        #pragma unroll
        for (int i = 0; i < 16; ++i) {
            int idx = tid + i * 256;
            int r = idx >> 5, c = idx & 31;
            xs[r][c] = X[(size_t)(rowBase + r) * DMODEL + (kb + c)];
        }
        // stage W tile transposed: wst[n][k] = W[kb+k][colBase+n]
        #pragma unroll
        for (int i = 0; i < 8; ++i) {
            int idx = tid + i * 256;
            int r = idx >> 6, c = idx & 63;        // r = k (0..31), c = n (0..63)
            wst[c][r] = Wsel[(size_t)(kb + r) * DMODEL + (colBase + c)];
        }
        // prefetch next k-block while WMMAs run (global_prefetch_b8)
        if (kb + 32 < DMODEL) {
            __builtin_prefetch(&X[(size_t)(rowBase + (tid >> 3)) * DMODEL + (kb + 32) + ((tid & 7) << 2)], 0, 0);
            __builtin_prefetch(&Wsel[(size_t)(kb + 32 + (tid >> 6)) * DMODEL + colBase + (tid & 63)], 0, 0);
        }
        __syncthreads();

        #pragma unroll
        for (int ks = 0; ks < 32; ks += 4) {
            // A fragment (16x4): lanes 0-15: M=l15,K={ks,ks+1}; lanes 16-31: K={ks+2,ks+3}
            v2f a  = *(const v2f*)&xs[waveRow + l15][ks + koff];
            // B fragments (4x16): single aligned b64 per lane from transposed tile
            v2f b0 = *(const v2f*)&wst[ 0 + l15][ks + koff];
            v2f b1 = *(const v2f*)&wst[16 + l15][ks + koff];
            v2f b2 = *(const v2f*)&wst[32 + l15][ks + koff];
            v2f b3 = *(const v2f*)&wst[48 + l15][ks + koff];

            acc0 = __builtin_amdgcn_wmma_f32_16x16x4_f32(false, a, false, b0, (short)0, acc0, false, false);
            acc1 = __builtin_amdgcn_wmma_f32_16x16x4_f32(false, a, false, b1, (short)0, acc1, false, false);
            acc2 = __builtin_amdgcn_wmma_f32_16x16x4_f32(false, a, false, b2, (short)0, acc2, false, false);
            acc3 = __builtin_amdgcn_wmma_f32_16x16x4_f32(false, a, false, b3, (short)0, acc3, false, false);
        }
    }

    // C/D layout: VGPR j -> row M=j (lanes 0-15) / M=j+8 (lanes 16-31), N=l15
    const int mAdd = (lane >> 4) << 3;
    #pragma unroll
    for (int j = 0; j < 8; ++j) {
        int row = rowBase + waveRow + j + mAdd;
        size_t o = (size_t)row * DMODEL + colBase;
        Osel[o +  0 + l15] = acc0[j];
        Osel[o + 16 + l15] = acc1[j];
        Osel[o + 32 + l15] = acc2[j];
        Osel[o + 48 + l15] = acc3[j];
    }
}

// ---------------------------------------------------------------------------
// Local attention: one wave32 per (b, s, h).
// Phase 1: lane j = window position j (j<31). Score = q . k_{s-15+j} / 8.
//          Zero-padded (out-of-range) keys -> score 0 (kept in softmax, as in
//          reference). Lane 31 (not a window slot) -> -inf.
// Phase 2: lanes switch to d-dimension (2 dims/lane), weights broadcast via
//          __shfl -> coalesced V loads and output stores.
// ---------------------------------------------------------------------------
__global__ __launch_bounds__(256)
void local_attn_kernel(const float* __restrict__ Q,
                       const float* __restrict__ Kp,
                       const float* __restrict__ Vp,
                       float* __restrict__ O)
{
    const int gwave = blockIdx.x * (blockDim.x >> 5) + (threadIdx.x >> 5);
    const int lane  = threadIdx.x & 31;
    const int h = gwave & (HEADS - 1);
    const int s = (gwave >> 3) & (SEQ - 1);
    const int b = gwave >> 15;              // / (HEADS*SEQ)

    const size_t rowOff = ((size_t)b * SEQ + s) * DMODEL + h * DH;

    const int j   = lane;
    const int src = s - HALFW + j;

    float score;
    if (j >= WIN) {
        score = -INFINITY;                  // lane 31: not a window slot
    } else if (src < 0 || src >= SEQ) {
        score = 0.0f;                       // zero-padded key -> zero dot
    } else {
        const float* qp = Q  + rowOff;
        const float* kp = Kp + ((size_t)b * SEQ + src) * DMODEL + h * DH;
        float dot = 0.0f;
        #pragma unroll
        for (int d = 0; d < DH; d += 4) {
            float4 qv = *(const float4*)(qp + d);
            float4 kv = *(const float4*)(kp + d);
            dot += qv.x * kv.x + qv.y * kv.y + qv.z * kv.z + qv.w * kv.w;
        }
        score = dot * 0.125f;               // 1/sqrt(64)
    }

    // wave32 softmax
    float m = score;
    #pragma unroll
    for (int off = 16; off > 0; off >>= 1) m = fmaxf(m, __shfl_xor(m, off, 32));
    float e = __expf(score - m);            // lane 31 -> exp(-inf) = 0
    float sum = e;
    #pragma unroll
    for (int off = 16; off > 0; off >>= 1) sum += __shfl_xor(sum, off, 32);
    const float a = e / sum;

    // A @ V: lane handles dims {lane, lane+32}
    float acc0 = 0.0f, acc1 = 0.0f;
    for (int jj = 0; jj < WIN; ++jj) {
        int sj = s - HALFW + jj;
        float aj = __shfl(a, jj, 32);
        if (sj >= 0 && sj < SEQ) {          // zero-padded value -> skip
            const float* vp = Vp + ((size_t)b * SEQ + sj) * DMODEL + h * DH;
            acc0 += aj * vp[lane];
            acc1 += aj * vp[lane + 32];
        }
    }
    O[rowOff + lane]      = acc0;
    O[rowOff + lane + 32] = acc1;
}

// ---------------------------------------------------------------------------
extern "C" void kernel_launch(void* const* d_in, const int* in_sizes, int n_in,
                              void* d_out, int out_size, void* d_ws, size_t ws_size,
                              hipStream_t stream)
{
    (void)in_sizes; (void)n_in; (void)out_size; (void)ws_size;
    const float* x  = (const float*)d_in[0];
    const float* Wq = (const float*)d_in[1];
    const float* Wk = (const float*)d_in[2];
    const float* Wv = (const float*)d_in[3];
    float* out = (float*)d_out;

    const size_t n = (size_t)BQ * SEQ * DMODEL;   // 4M floats per tensor
    float* qb = (float*)d_ws;
    float* kb = qb + n;
    float* vb = kb + n;

    dim3 gProj(BQ * SEQ / 128, DMODEL / 64, 3);   // (64, 8, 3)
    proj_gemm_kernel<<<gProj, 256, 0, stream>>>(x, Wq, Wk, Wv, qb, kb, vb);

    const int totalWaves = BQ * SEQ * HEADS;      // 65536
    local_attn_kernel<<<totalWaves / 8, 256, 0, stream>>>(qb, kb, vb, out);
}